// Convolution_29738353557732
// MI455X (gfx1250) — compile-verified
//
#include <hip/hip_runtime.h>
#include <math.h>

typedef __attribute__((ext_vector_type(2))) float v2f;
typedef __attribute__((ext_vector_type(8))) float v8f;

#define INV_SQRT8  0.35355339059327373f
#define INV_SQRT32 0.17677669529663687f
#define INV_SQRT64 0.125f
#define INV_SQRT96 0.10206207261596575f
#define INV3F      0.57735026918962576f

// D = A(16x4,f32) * B(4x16,f32) + C(16x16,f32) on the CDNA5 matrix pipe.
static __device__ __forceinline__ v8f wmma_f32(v2f a, v2f b, v8f c) {
  return __builtin_amdgcn_wmma_f32_16x16x4_f32(false, a, false, b, (short)0, c,
                                               false, false);
}
static __device__ __forceinline__ v8f vzero() {
  v8f z = {0.f, 0.f, 0.f, 0.f, 0.f, 0.f, 0.f, 0.f};
  return z;
}
// A fragment (ISA 7.12.2): lane L holds row (L&15), K pair at 4*ks + 2*(L>>4).
static __device__ __forceinline__ v2f load_a(const float* __restrict__ A, int lda,
                                             int r0, int ks, int l15, int half) {
  const float* p = A + (size_t)(r0 + l15) * lda + 4 * ks + 2 * half;
  v2f a; a.x = p[0]; a.y = p[1]; return a;
}
// A fragment with element stride 3 (interleaved xyz vector channels).
static __device__ __forceinline__ v2f load_a3(const float* __restrict__ A, int lda,
                                              int r0, int ks, int l15, int half) {
  const float* p = A + (size_t)(r0 + l15) * lda + (size_t)(4 * ks + 2 * half) * 3;
  v2f a; a.x = p[0]; a.y = p[3]; return a;
}
// B fragment: lane L holds col (L&15), rows 4*ks+2*(L>>4) and +1.
static __device__ __forceinline__ v2f load_b(const float* __restrict__ B, int ldb,
                                             int c0, int ks, int l15, int half) {
  const float* p = B + (size_t)(4 * ks + 2 * half) * ldb + c0 + l15;
  v2f b; b.x = p[0]; b.y = p[ldb]; return b;
}

// ---------------------------------------------------------------- zero / pack
__global__ void k_zero(float* __restrict__ p, size_t n) {
  size_t i = (size_t)blockIdx.x * blockDim.x + threadIdx.x;
  size_t st = (size_t)gridDim.x * blockDim.x;
  for (; i < n; i += st) p[i] = 0.f;
}
// Pack fc_w2 (64x192) as consecutive-row pairs so B fragments load as one b64.
__global__ void k_pack_w2(const float* __restrict__ w2, float* __restrict__ w2p) {
  int i = blockIdx.x * 256 + threadIdx.x;            // 0 .. 32*192-1
  if (i >= 32 * 192) return;
  int r2 = i / 192, col = i % 192;
  w2p[2 * i + 0] = w2[(2 * r2 + 0) * 192 + col];
  w2p[2 * i + 1] = w2[(2 * r2 + 1) * 192 + col];
}

// --------------------------------------------------- node lin1 -> f_s / f_v
__global__ void __launch_bounds__(32) k_node_lin1(
    const float* __restrict__ node_input, const float* __restrict__ node_attr,
    const float* __restrict__ w0, const float* __restrict__ w1,
    float* __restrict__ f_s, float* __restrict__ f_v, int N) {
  int l = threadIdx.x & 31, l15 = l & 15, half = l >> 4;
  int r0 = blockIdx.x * 16;
  if (r0 + 16 > N) return;

  v8f acc[4];
#pragma unroll
  for (int nt = 0; nt < 4; ++nt) acc[nt] = vzero();
#pragma unroll
  for (int k = 0; k < 16; ++k) {
    v2f a = load_a(node_input, 160, r0, k, l15, half);
#pragma unroll
    for (int nt = 0; nt < 4; ++nt)
      acc[nt] = wmma_f32(a, load_b(w0, 64, nt * 16, k, l15, half), acc[nt]);
  }
#pragma unroll
  for (int j = 0; j < 8; ++j) {
    int m = j + 8 * half;
    float at = node_attr[r0 + m] * INV_SQRT64;
#pragma unroll
    for (int nt = 0; nt < 4; ++nt)
      f_s[(size_t)(r0 + m) * 64 + nt * 16 + l15] = acc[nt][j] * at;
  }
#pragma unroll
  for (int d = 0; d < 3; ++d) {
    v8f av[2]; av[0] = vzero(); av[1] = vzero();
#pragma unroll
    for (int k = 0; k < 8; ++k) {
      v2f a = load_a3(node_input + 64 + d, 160, r0, k, l15, half);
#pragma unroll
      for (int nt = 0; nt < 2; ++nt)
        av[nt] = wmma_f32(a, load_b(w1, 32, nt * 16, k, l15, half), av[nt]);
    }
#pragma unroll
    for (int j = 0; j < 8; ++j) {
      int m = j + 8 * half;
      float at = node_attr[r0 + m] * INV_SQRT32;
#pragma unroll
      for (int nt = 0; nt < 2; ++nt)
        f_v[(size_t)(r0 + m) * 96 + (nt * 16 + l15) * 3 + d] = av[nt][j] * at;
    }
  }
}

// -------------------------------------- fused edge MLP + tensor-product + scatter
__global__ void __launch_bounds__(32) k_edge(
    const float* __restrict__ edge_scalars, const float* __restrict__ edge_attr,
    const int* __restrict__ edge_src, const int* __restrict__ edge_dst,
    const float* __restrict__ fc_w1, const float* __restrict__ fc_w2p,
    const float* __restrict__ f_s, const float* __restrict__ f_v,
    float* __restrict__ mid, int E) {
  __shared__ float lds_h[16 * 68];  // padded stride 68 -> conflict-free A reads
  int l = threadIdx.x & 31, l15 = l & 15, half = l >> 4;
  int e0 = blockIdx.x * 16;
  if (e0 + 16 > E) return;

  // GEMM1: h = silu(es @ fc_w1 / sqrt(8)); fold the later 1/sqrt(64) into h.
  v8f h[4];
#pragma unroll
  for (int nt = 0; nt < 4; ++nt) h[nt] = vzero();
#pragma unroll
  for (int k = 0; k < 2; ++k) {
    v2f a = load_a(edge_scalars, 8, e0, k, l15, half);
#pragma unroll
    for (int nt = 0; nt < 4; ++nt)
      h[nt] = wmma_f32(a, load_b(fc_w1, 64, nt * 16, k, l15, half), h[nt]);
  }
#pragma unroll
  for (int nt = 0; nt < 4; ++nt)
#pragma unroll
    for (int j = 0; j < 8; ++j) {
      float x = h[nt][j] * INV_SQRT8;
      x = x / (1.f + __expf(-x));                    // silu
      lds_h[(j + 8 * half) * 68 + nt * 16 + l15] = x * INV_SQRT64;
    }
  __syncthreads();

  // GEMM2: weight(16x192) = h @ fc_w2 (A from LDS, B packed row-pairs).
  v8f w[12];
#pragma unroll
  for (int nt = 0; nt < 12; ++nt) w[nt] = vzero();
#pragma unroll
  for (int k = 0; k < 16; ++k) {
    int kk = 4 * k + 2 * half;
    v2f a; a.x = lds_h[l15 * 68 + kk]; a.y = lds_h[l15 * 68 + kk + 1];
    const float* bp = fc_w2p + (size_t)(kk >> 1) * 384;
#pragma unroll
    for (int nt = 0; nt < 12; ++nt) {
      v2f b = *(const v2f*)(bp + (nt * 16 + l15) * 2);
      w[nt] = wmma_f32(a, b, w[nt]);
    }
  }

  // Tensor product + scatter-add into L2-resident mid (dst-indexed).
#pragma unroll
  for (int j = 0; j < 8; ++j) {
    int e = e0 + j + 8 * half;                        // edge owned by this VGPR row
    int src = edge_src[e], dst = edge_dst[e];
    float4 ea = *(const float4*)(edge_attr + (size_t)e * 4);
    float ys = ea.x, y0 = ea.y, y1 = ea.z, y2 = ea.w;
    const float* fs = f_s + (size_t)src * 64;
    const float* fv = f_v + (size_t)src * 96;
    float* md = mid + (size_t)dst * 384;
    int n = l15;
    float g0 = fs[n], g1 = fs[16 + n], g2 = fs[32 + n], g3 = fs[48 + n];
    // m0a (cols 0..63, w1_)
    atomicAdd(md + n,      g0 * ys * w[0][j]);
    atomicAdd(md + 16 + n, g1 * ys * w[1][j]);
    atomicAdd(md + 32 + n, g2 * ys * w[2][j]);
    atomicAdd(md + 48 + n, g3 * ys * w[3][j]);
    // m1a (cols 96 + u*3 + d, w2_)
    { float s = g0 * w[4][j]; int c = 96 + n * 3;
      atomicAdd(md + c, s * y0); atomicAdd(md + c + 1, s * y1); atomicAdd(md + c + 2, s * y2); }
    { float s = g1 * w[5][j]; int c = 96 + (16 + n) * 3;
      atomicAdd(md + c, s * y0); atomicAdd(md + c + 1, s * y1); atomicAdd(md + c + 2, s * y2); }
    { float s = g2 * w[6][j]; int c = 96 + (32 + n) * 3;
      atomicAdd(md + c, s * y0); atomicAdd(md + c + 1, s * y1); atomicAdd(md + c + 2, s * y2); }
    { float s = g3 * w[7][j]; int c = 96 + (48 + n) * 3;
      atomicAdd(md + c, s * y0); atomicAdd(md + c + 1, s * y1); atomicAdd(md + c + 2, s * y2); }
    // gv rows w = n and 16+n
    float va0 = fv[n * 3], va1 = fv[n * 3 + 1], va2 = fv[n * 3 + 2];
    float vb0 = fv[(16 + n) * 3], vb1 = fv[(16 + n) * 3 + 1], vb2 = fv[(16 + n) * 3 + 2];
    // m1b (cols 288 + w*3 + d, w3_)
    { float s = ys * w[8][j]; int c = 288 + n * 3;
      atomicAdd(md + c, va0 * s); atomicAdd(md + c + 1, va1 * s); atomicAdd(md + c + 2, va2 * s); }
    { float s = ys * w[9][j]; int c = 288 + (16 + n) * 3;
      atomicAdd(md + c, vb0 * s); atomicAdd(md + c + 1, vb1 * s); atomicAdd(md + c + 2, vb2 * s); }
    // m0b (cols 64..95, w4_)
    atomicAdd(md + 64 + n,      (va0 * y0 + va1 * y1 + va2 * y2) * INV3F * w[10][j]);
    atomicAdd(md + 64 + 16 + n, (vb0 * y0 + vb1 * y1 + vb2 * y2) * INV3F * w[11][j]);
  }
}

// ------------------------------------------- node: sc + lin2 + angle -> output
__global__ void __launch_bounds__(32) k_out(
    const float* __restrict__ node_input, const float* __restrict__ node_attr,
    const float* __restrict__ sc_w0, const float* __restrict__ sc_w1,
    const float* __restrict__ lin2_w0, const float* __restrict__ lin2_w1,
    const float* __restrict__ lin3_w, const float* __restrict__ mid,
    const int* __restrict__ num_neighbors, float* __restrict__ out, int N) {
  __shared__ float lds_cs[16], lds_sn[16];
  int l = threadIdx.x & 31, l15 = l & 15, half = l >> 4;
  int r0 = blockIdx.x * 16;
  if (r0 + 16 > N) return;
  float sc2 = rsqrtf((float)num_neighbors[0]);       // 1/sqrt(num_neighbors)

  // angle = 0.1 * (mid_s @ lin3_w) / sqrt(96) * attr  (mid scaled by sc2)
  {
    const float* ms = mid + (size_t)(r0 + l15) * 384 + half * 48;
    const float* lw = lin3_w + half * 48;
    float partial = 0.f;
#pragma unroll 8
    for (int c = 0; c < 48; ++c) partial = fmaf(ms[c], lw[c], partial);
    partial += __shfl_xor(partial, 16, 32);
    float ang = 0.1f * partial * INV_SQRT96 * sc2 * node_attr[r0 + l15];
    if (l < 16) { lds_cs[l15] = cosf(ang); lds_sn[l15] = sinf(ang); }
  }
  __syncthreads();

  // scalar section (output cols 0..63)
  v8f ss[4], cs_[4];
#pragma unroll
  for (int nt = 0; nt < 4; ++nt) { ss[nt] = vzero(); cs_[nt] = vzero(); }
#pragma unroll
  for (int k = 0; k < 16; ++k) {
    v2f a = load_a(node_input, 160, r0, k, l15, half);
#pragma unroll
    for (int nt = 0; nt < 4; ++nt)
      ss[nt] = wmma_f32(a, load_b(sc_w0, 64, nt * 16, k, l15, half), ss[nt]);
  }
#pragma unroll
  for (int k = 0; k < 24; ++k) {
    v2f a = load_a(mid, 384, r0, k, l15, half);
#pragma unroll
    for (int nt = 0; nt < 4; ++nt)
      cs_[nt] = wmma_f32(a, load_b(lin2_w0, 64, nt * 16, k, l15, half), cs_[nt]);
  }
#pragma unroll
  for (int j = 0; j < 8; ++j) {
    int m = j + 8 * half, node = r0 + m;
    float at = node_attr[node];
    float cA = lds_cs[m], sA = lds_sn[m];
#pragma unroll
    for (int nt = 0; nt < 4; ++nt)
      out[(size_t)node * 160 + nt * 16 + l15] =
          cA * (ss[nt][j] * INV_SQRT64 * at) +
          sA * (cs_[nt][j] * INV_SQRT96 * sc2 * at);
  }

  // vector section (output cols 64 + w*3 + d)
#pragma unroll
  for (int d = 0; d < 3; ++d) {
    v8f sv[2], cv[2];
#pragma unroll
    for (int nt = 0; nt < 2; ++nt) { sv[nt] = vzero(); cv[nt] = vzero(); }
#pragma unroll
    for (int k = 0; k < 8; ++k) {
      v2f a = load_a3(node_input + 64 + d, 160, r0, k, l15, half);
#pragma unroll
      for (int nt = 0; nt < 2; ++nt)
        sv[nt] = wmma_f32(a, load_b(sc_w1, 32, nt * 16, k, l15, half), sv[nt]);
    }
#pragma unroll
    for (int k = 0; k < 24; ++k) {
      v2f a = load_a3(mid + 96 + d, 384, r0, k, l15, half);
#pragma unroll
      for (int nt = 0; nt < 2; ++nt)
        cv[nt] = wmma_f32(a, load_b(lin2_w1, 32, nt * 16, k, l15, half), cv[nt]);
    }
#pragma unroll
    for (int j = 0; j < 8; ++j) {
      int m = j + 8 * half, node = r0 + m;
      float at = node_attr[node];
      float cA = lds_cs[m], sA = lds_sn[m];
#pragma unroll
      for (int nt = 0; nt < 2; ++nt)
        out[(size_t)node * 160 + 64 + (nt * 16 + l15) * 3 + d] =
            cA * (sv[nt][j] * INV_SQRT32 * at) +
            sA * (cv[nt][j] * INV_SQRT96 * sc2 * at);
    }
  }
}

extern "C" void kernel_launch(void* const* d_in, const int* in_sizes, int n_in,
                              void* d_out, int out_size, void* d_ws, size_t ws_size,
                              hipStream_t stream) {
  const float* node_input   = (const float*)d_in[0];
  const float* node_attr    = (const float*)d_in[1];
  const float* edge_attr    = (const float*)d_in[2];
  const float* edge_scalars = (const float*)d_in[3];
  const float* sc_w0        = (const float*)d_in[4];
  const float* sc_w1        = (const float*)d_in[5];
  const float* lin1_w0      = (const float*)d_in[6];
  const float* lin1_w1      = (const float*)d_in[7];
  const float* fc_w1        = (const float*)d_in[8];
  const float* fc_w2        = (const float*)d_in[9];
  const float* lin2_w0      = (const float*)d_in[10];
  const float* lin2_w1      = (const float*)d_in[11];
  const float* lin3_w       = (const float*)d_in[12];
  const int*   edge_src     = (const int*)d_in[13];
  const int*   edge_dst     = (const int*)d_in[14];
  const int*   num_neighbors= (const int*)d_in[15];
  float* out = (float*)d_out;

  int N = in_sizes[0] / 160;
  int E = in_sizes[2] / 4;

  float* ws  = (float*)d_ws;
  float* f_s = ws;                                   // N*64
  float* f_v = ws + (size_t)N * 64;                  // N*96
  float* mid = ws + (size_t)N * 160;                 // N*384 (segment sum)
  float* w2p = ws + (size_t)N * 544;                 // 64*192 packed fc_w2

  k_zero<<<2048, 256, 0, stream>>>(mid, (size_t)N * 384);
  k_pack_w2<<<(32 * 192 + 255) / 256, 256, 0, stream>>>(fc_w2, w2p);
  k_node_lin1<<<(N + 15) / 16, 32, 0, stream>>>(node_input, node_attr,
                                                lin1_w0, lin1_w1, f_s, f_v, N);
  k_edge<<<(E + 15) / 16, 32, 0, stream>>>(edge_scalars, edge_attr, edge_src,
                                           edge_dst, fc_w1, w2p, f_s, f_v, mid, E);
  k_out<<<(N + 15) / 16, 32, 0, stream>>>(node_input, node_attr, sc_w0, sc_w1,
                                          lin2_w0, lin2_w1, lin3_w, mid,
                                          num_neighbors, out, N);
}